// VarianceLayer_63050119906000
// MI455X (gfx1250) — compile-verified
//
#include <hip/hip_runtime.h>
#include <stdint.h>

// CDNA5 / gfx1250: wave32, WMMA, async global->LDS copies.
typedef float v2f __attribute__((ext_vector_type(2)));
typedef float v8f __attribute__((ext_vector_type(8)));

#define ROWS     8          // image rows per tile-row strip
#define CHUNK_W  768        // floats per image row per chunk (32 tiles * 8 px * 3 ch)
#define NCHUNK   4          // 128 tiles / 32 tiles per chunk
#define THREADS  192        // 6 waves of 32

// Issue one chunk's async copies: 8 rows x 3072 B, each thread moves one 16B
// unit per row (192 threads * 16B = 3072B row), 512B contiguous per wave-instr.
__device__ __forceinline__ void issue_chunk_async(const float* gsrc_row0,
                                                  const float* lds_row0,
                                                  int tid)
{
    unsigned lds0 = (unsigned)(uintptr_t)lds_row0 + (unsigned)tid * 16u;
    uint64_t g0   = (uint64_t)(uintptr_t)gsrc_row0 + (uint64_t)tid * 16u;
#pragma unroll
    for (int r = 0; r < ROWS; ++r) {
        asm volatile("global_load_async_to_lds_b128 %0, %1, off"
                     :: "v"(lds0 + (unsigned)(r * 3072)),   // LDS row stride 768 f32
                        "v"(g0 + (uint64_t)r * 12288u)      // global row stride 1024*3*4 B
                     : "memory");
    }
}

__global__ __launch_bounds__(THREADS)
void tile_var8x8_kernel(const float* __restrict__ x, float* __restrict__ out)
{
    __shared__ float smem[2][ROWS * CHUNK_W];   // 2 x 24 KB, double buffered

    const int tid = threadIdx.x;
    const int ty  = blockIdx.x;     // tile row 0..127
    const int b   = blockIdx.y;     // batch    0..15

    // base of this strip: x[b][ty*8][0][0]; row stride = 3072 floats
    const float* strip = x + (size_t)b * (1024u * 1024u * 3u)
                           + (size_t)ty * 8u * 3072u;

    const int lane = tid & 31;
    const int w    = tid >> 5;          // wave 0..5
    const int j    = lane & 15;         // WMMA column 0..15
    const int h    = lane >> 4;         // lane half -> rows 0-3 vs 4-7
    const int o    = w * 16 + j;        // output index in chunk, 0..95
    const int tile = o / 3;             // tile-in-chunk 0..31
    const int c    = o % 3;             // channel

    // prime the pipeline with chunk 0
    issue_chunk_async(strip, &smem[0][0], tid);

    for (int g = 0; g < NCHUNK; ++g) {
        const float* cur = &smem[g & 1][0];

        if (g + 1 < NCHUNK) {
            // chunk g+1 starts 768 floats (3072 B) further along each row
            issue_chunk_async(strip + (size_t)(g + 1) * CHUNK_W,
                              &smem[(g + 1) & 1][0], tid);
            // keep only the next chunk's 8 transfers outstanding
            asm volatile("s_wait_asynccnt 8" ::: "memory");
        } else {
            asm volatile("s_wait_asynccnt 0" ::: "memory");
        }
        __syncthreads();   // chunk g visible to all waves

        // Per-lane serial accumulation: 2 tile-rows x 8 cols per B slot.
        // Lane half h covers tile rows 4h..4h+3.
        const float* basep = cur + (h * 4) * CHUNK_W + tile * 24 + c;
        float s0 = 0.f, s1 = 0.f, q0 = 0.f, q1 = 0.f;
#pragma unroll
        for (int rr = 0; rr < 2; ++rr)
#pragma unroll
            for (int q = 0; q < 8; ++q) {
                float v = basep[rr * CHUNK_W + q * 3];
                s0 += v;
                q0 = fmaf(v, v, q0);
            }
#pragma unroll
        for (int rr = 2; rr < 4; ++rr)
#pragma unroll
            for (int q = 0; q < 8; ++q) {
                float v = basep[rr * CHUNK_W + q * 3];
                s1 += v;
                q1 = fmaf(v, v, q1);
            }

        // Cross-slot reduction via WMMA: D = ones(16x4) * B(4x16) + 0
        // => every row of D holds the column sums; column j lives in lanes
        // {j, j+16}, so d[0] on lane j is the full 64-element reduction.
        v2f ones; ones[0] = 1.0f; ones[1] = 1.0f;
        v2f bs;   bs[0] = s0;     bs[1] = s1;
        v2f bq;   bq[0] = q0;     bq[1] = q1;
        v8f z = {};
        v8f dsum = __builtin_amdgcn_wmma_f32_16x16x4_f32(
            false, ones, false, bs, (short)0, z, false, false);
        v8f dsq  = __builtin_amdgcn_wmma_f32_16x16x4_f32(
            false, ones, false, bq, (short)0, z, false, false);

        if (h == 0) {
            float sum  = dsum[0];
            float sq   = dsq[0];
            float mean = sum * 0.015625f;                  // /64
            int tx = g * 32 + tile;
            out[(((size_t)b * 128 + ty) * 128 + tx) * 3 + c] =
                fmaf(-mean, mean, sq * 0.015625f);         // E[x^2]-E[x]^2
        }
        __syncthreads();   // all reads of this buffer done before its reuse
    }
}

extern "C" void kernel_launch(void* const* d_in, const int* in_sizes, int n_in,
                              void* d_out, int out_size, void* d_ws, size_t ws_size,
                              hipStream_t stream)
{
    (void)in_sizes; (void)n_in; (void)d_ws; (void)ws_size; (void)out_size;
    const float* x = (const float*)d_in[0];
    float* out = (float*)d_out;

    dim3 grid(128, 16);   // (tile rows, batch)
    tile_var8x8_kernel<<<grid, THREADS, 0, stream>>>(x, out);
}